// EncodingLayer_77326591197578
// MI455X (gfx1250) — compile-verified
//
#include <hip/hip_runtime.h>
#include <stdint.h>

// Problem sizes (fixed by the reference setup).
#define Bn 16
#define Sn 64
#define In 128
#define Hn 512
#define Tn 5
#define ALPHA_F 0.9f

// Phase-1 tiling: block = one (b,s), 128 threads, each owns 4 consecutive h.
#define THREADS1 128
#define CHUNK_I  16                         // i-rows staged per LDS buffer
#define NCHUNK   (In / CHUNK_I)             // 8 chunks
#define ROW_BYTES (Hn * 4)                  // 2048: global i-row stride
#define LDS_ROW   (THREADS1 * 16)           // 2048: LDS row stride (== ROW_BYTES!)
#define CHUNK_BYTES (CHUNK_I * ROW_BYTES)   // 32768 per buffer

// GVS-mode async copy of CHUNK_I rows: global = sbase + voff + imm,
// lds = lds_dst + imm. Row stride 2048 matches on both sides, so the
// shared 24-bit INST_OFFSET immediate addresses global AND LDS rows —
// zero per-load VALU address math. x is streamed once -> TH_LOAD_NT.
__device__ __forceinline__ void async_chunk_to_lds(uint32_t lds_dst, uint32_t voff,
                                                   uint64_t sbase) {
#pragma unroll
    for (int il = 0; il < CHUNK_I; ++il) {
        asm volatile("global_load_async_to_lds_b128 %0, %1, %2 offset:%3 th:TH_LOAD_NT"
                     :: "v"(lds_dst), "v"(voff), "s"(sbase), "n"(il * ROW_BYTES)
                     : "memory");
    }
}

// cur[bs,h] = sum_i x[bs,i,h] * enc[i,h]
__global__ __launch_bounds__(THREADS1)
void einsum_ih_kernel(const float* __restrict__ x,
                      const float* __restrict__ enc,
                      float* __restrict__ cur) {
    __shared__ float4 buf[2][CHUNK_I][THREADS1];   // 2 x 32KB = 64KB LDS
    const int bs = blockIdx.x;                     // b*Sn + s
    const int t  = threadIdx.x;                    // h4 index

    // Wave-uniform global base for this (b,s); chunk advance stays scalar (SALU).
    const float*   xblk = x + (size_t)bs * In * Hn;
    const uint32_t voff = (uint32_t)(t * 16);      // per-lane h4 byte offset
    const uint32_t lds_t = (uint32_t)(uintptr_t)(&buf[0][0][0])
                         + (uint32_t)(t * sizeof(float4));
    const uint32_t lds_dst[2] = { lds_t, lds_t + CHUNK_BYTES };

    // Prefetch chunk 0.
    async_chunk_to_lds(lds_dst[0], voff, (uint64_t)(uintptr_t)xblk);

    float4 acc = make_float4(0.f, 0.f, 0.f, 0.f);

    for (int c = 0; c < NCHUNK; ++c) {
        if (c + 1 < NCHUNK) {
            // Issue next chunk, then wait until only those 16 remain in flight
            // (async loads complete in order -> oldest 16 == chunk c done).
            const uint64_t sb = (uint64_t)(uintptr_t)(xblk + (size_t)(c + 1) * CHUNK_I * Hn);
            async_chunk_to_lds(lds_dst[(c + 1) & 1], voff, sb);
            asm volatile("s_wait_asynccnt 0x10" ::: "memory");
        } else {
            asm volatile("s_wait_asynccnt 0x0" ::: "memory");   // drain
        }

        const int    bsel = c & 1;
        const float* e    = enc + (size_t)c * CHUNK_I * Hn + 4 * t;
#pragma unroll
        for (int il = 0; il < CHUNK_I; ++il) {
            float4 xv = buf[bsel][il][t];
            float4 ev = *reinterpret_cast<const float4*>(e);
            acc.x = fmaf(xv.x, ev.x, acc.x);
            acc.y = fmaf(xv.y, ev.y, acc.y);
            acc.z = fmaf(xv.z, ev.z, acc.z);
            acc.w = fmaf(xv.w, ev.w, acc.w);
            e += Hn;
        }
    }

    *reinterpret_cast<float4*>(cur + (size_t)bs * Hn + 4 * t) = acc;
}

// LIF recurrence per (b,h) lane; strictly sequential over S*T steps.
// Output is write-once streaming -> nontemporal stores keep L2 for cur/enc.
__global__ __launch_bounds__(256)
void lif_scan_kernel(const float* __restrict__ cur, float* __restrict__ out) {
    const int idx = blockIdx.x * 256 + threadIdx.x;   // 0 .. B*H-1
    const int b = idx / Hn;
    const int h = idx % Hn;

    const float* c = cur + (size_t)b * Sn * Hn + h;
    float*       o = out + (size_t)b * Sn * Tn * Hn + h;

    float v = 0.f;
    for (int s = 0; s < Sn; ++s) {
        const float it = c[(size_t)s * Hn];
#pragma unroll
        for (int tt = 0; tt < Tn; ++tt) {
            v = ALPHA_F * v + it;
            const float z = (v > 1.0f) ? 1.0f : 0.0f;  // spike(v - 1)
            v -= z;                                     // soft reset (THRESHOLD=1)
            __builtin_nontemporal_store(z, &o[(size_t)(s * Tn + tt) * Hn]);
        }
    }
}

extern "C" void kernel_launch(void* const* d_in, const int* in_sizes, int n_in,
                              void* d_out, int out_size, void* d_ws, size_t ws_size,
                              hipStream_t stream) {
    (void)in_sizes; (void)n_in; (void)out_size; (void)ws_size;

    const float* x   = (const float*)d_in[0];
    const float* enc = (const float*)d_in[1];
    float*       out = (float*)d_out;
    float*       cur = (float*)d_ws;     // Bn*Sn*Hn floats = 2 MB scratch

    einsum_ih_kernel<<<dim3(Bn * Sn), dim3(THREADS1), 0, stream>>>(x, enc, cur);
    lif_scan_kernel<<<dim3((Bn * Hn) / 256), dim3(256), 0, stream>>>(cur, out);
}